// CIAM_30090540876359
// MI455X (gfx1250) — compile-verified
//
#include <hip/hip_runtime.h>

typedef __attribute__((ext_vector_type(2))) float v2f;
typedef __attribute__((ext_vector_type(4))) float v4f;
typedef __attribute__((ext_vector_type(8))) float v8f;

#define C_DIM 64
#define HH 256
#define WW 256
#define S 4
#define PATCHES_PER_BLK 16
#define CH_STRIDE 66     // padded row stride for [16][64] LDS arrays (bank-conflict avoidance)
#define OUT_PSTRIDE 20   // pixel pitch for output staging: 20 floats = 80B (16B aligned, gcd(20,64)=4)

__device__ __forceinline__ float sigmoidf_(float x) {
    return 1.0f / (1.0f + __expf(-x));
}

__global__ __launch_bounds__(256)
void ciam_fused_kernel(const float* __restrict__ x,
                       const float* __restrict__ fc_w,
                       const float* __restrict__ fc_b,
                       const float* __restrict__ conv1_w,
                       const float* __restrict__ conv1_b,
                       const float* __restrict__ conv2_w,
                       const float* __restrict__ conv2_b,
                       float* __restrict__ out)
{
    __shared__ float chmax_s[PATCHES_PER_BLK * CH_STRIDE];
    __shared__ float m_s[PATCHES_PER_BLK * CH_STRIDE];
    __shared__ __align__(16) float lds_out[16 * PATCHES_PER_BLK * OUT_PSTRIDE]; // 20 KB

    const int tid = threadIdx.x;
    // block -> (b, patch-row ph, group of 16 patches along W)
    const int pwg = blockIdx.x & 3;          // 4 groups of 16 patches along W (64 patches/row)
    const int ph  = (blockIdx.x >> 2) & 63;
    const int b   = blockIdx.x >> 8;

    const long long HW = (long long)HH * WW;
    const float* xb = x + (long long)b * C_DIM * HW;
    float*       ob = out + (long long)b * C_DIM * HW;
    const int col0 = pwg * (PATCHES_PER_BLK * S);
    const int row0 = ph * S;

    // ---------- Phase 1: per-(patch, channel) max over the 16 patch pixels ----------
    // thread handles 4 (patch, channel) pairs; float4 loads are coalesced across lanes.
    #pragma unroll
    for (int k = 0; k < 4; ++k) {
        int i = k * 256 + tid;       // 1024 (patch, c) pairs total
        int patch = i & 15;
        int c = i >> 4;              // 0..63
        const float* base = xb + (long long)c * HW + (long long)row0 * WW + col0 + patch * S;
        float mx = -__builtin_inff();
        #pragma unroll
        for (int r = 0; r < S; ++r) {
            v4f v = *(const v4f*)(base + (long long)r * WW);
            mx = fmaxf(mx, fmaxf(fmaxf(v.x, v.y), fmaxf(v.z, v.w)));
        }
        chmax_s[patch * CH_STRIDE + c] = mx;
    }
    __syncthreads();

    // ---------- GEMM via V_WMMA_F32_16X16X4_F32: m = sigmoid(chmax @ fc_w^T + fc_b) ----------
    const int wave = tid >> 5;
    const int lane = tid & 31;
    if (wave < 4) {
        const int nl    = lane & 15;
        const int khalf = lane >> 4;           // 0: K{0,1}, 1: K{2,3} per ISA A/B layout
        const int n     = wave * 16 + nl;      // output channel (N)
        const float bias = fc_b[n];
        v8f acc = {bias, bias, bias, bias, bias, bias, bias, bias};
        #pragma unroll
        for (int kk = 0; kk < 16; ++kk) {      // K = 64 in steps of 4
            const int kb = kk * 4 + khalf * 2;
            v2f a, bf;
            a.x  = chmax_s[nl * CH_STRIDE + kb];        // A[M=nl][K=kb]
            a.y  = chmax_s[nl * CH_STRIDE + kb + 1];    // A[M=nl][K=kb+1]
            bf.x = fc_w[n * C_DIM + kb];                // B[K=kb][N=n] = fc_w[n][kb]
            bf.y = fc_w[n * C_DIM + kb + 1];
            acc = __builtin_amdgcn_wmma_f32_16x16x4_f32(
                false, a, false, bf, (short)0, acc, false, false);
        }
        #pragma unroll
        for (int j = 0; j < 8; ++j) {          // D VGPR j holds rows j (lanes 0-15) / j+8 (16-31)
            const int rowm = j + 8 * khalf;
            m_s[rowm * CH_STRIDE + n] = sigmoidf_(acc[j]);
        }
    }
    __syncthreads();

    // ---------- Phase 2: per-pixel thread keeps all 64 channels in registers ----------
    const float c1w0 = conv1_w[0], c1w1 = conv1_w[1], c1b = conv1_b[0];
    const float c2w0 = conv2_w[0], c2w1 = conv2_w[1], c2b = conv2_b[0];

    const int patch = tid >> 4;                // lanes 0-15 / 16-31 are two patches per wave
    const int px    = tid & 15;
    const int pr    = px >> 2, pc = px & 3;
    const float* pbase = xb + (long long)(row0 + pr) * WW + col0 + patch * S + pc;

    float v[C_DIM];
    float sum = 0.f, mx = -__builtin_inff();
    #pragma unroll
    for (int c = 0; c < C_DIM; ++c) {          // re-read is L2-resident (134MB < 192MB L2); last use
        float t = __builtin_nontemporal_load(pbase + (long long)c * HW)
                  * m_s[patch * CH_STRIDE + c];
        v[c] = t;
        sum += t;
        mx = fmaxf(mx, t);
    }
    const float g1 = sigmoidf_(c1w0 * (sum * (1.0f / 64.0f)) + c1w1 * mx + c1b);

    float sum2 = 0.f, mx2 = -__builtin_inff();
    #pragma unroll
    for (int c = 0; c < C_DIM; ++c) {
        float t = v[c] * g1;
        v[c] = t;
        sum2 += t;
        mx2 = fmaxf(mx2, t);
    }
    // reduce over the 16 lanes of this patch (xor masks < 16 stay within the half-wave)
    #pragma unroll
    for (int o = 1; o < 16; o <<= 1) {
        sum2 += __shfl_xor(sum2, o, 32);
        mx2 = fmaxf(mx2, __shfl_xor(mx2, o, 32));
    }
    const float g2 = sigmoidf_(c2w0 * (sum2 * (1.0f / 1024.0f)) + c2w1 * mx2 + c2b);

    #pragma unroll
    for (int c = 0; c < C_DIM; ++c) {
        v[c] *= g2;
    }

    // ---------- Output: stage 16-channel chunks through LDS, emit coalesced B128 NT stores ----
    const int s_cl    = tid >> 4;   // store-phase channel-local index (0..15)
    const int s_patch = tid & 15;   // store-phase patch index
    #pragma unroll
    for (int chunk = 0; chunk < 4; ++chunk) {
        const int cbase = chunk * 16;
        __syncthreads();            // previous chunk's readers are done with lds_out
        #pragma unroll
        for (int cl = 0; cl < 16; ++cl) {   // writer mapping: thread = (patch, px)
            lds_out[(cl * PATCHES_PER_BLK + patch) * OUT_PSTRIDE + px] = v[cbase + cl];
        }
        __syncthreads();
        // reader mapping: thread = (channel-local, patch); rows give 4 coalesced B128 stores
        const float* src = &lds_out[(s_cl * PATCHES_PER_BLK + s_patch) * OUT_PSTRIDE];
        float* dst = ob + (long long)(cbase + s_cl) * HW
                        + (long long)row0 * WW + col0 + s_patch * S;
        #pragma unroll
        for (int r = 0; r < S; ++r) {
            v4f t = *(const v4f*)(src + r * 4);
            __builtin_nontemporal_store(t, (v4f*)(dst + (long long)r * WW));
        }
    }
}

extern "C" void kernel_launch(void* const* d_in, const int* in_sizes, int n_in,
                              void* d_out, int out_size, void* d_ws, size_t ws_size,
                              hipStream_t stream) {
    (void)in_sizes; (void)n_in; (void)d_ws; (void)ws_size; (void)out_size;
    const float* x       = (const float*)d_in[0];
    const float* fc_w    = (const float*)d_in[1];
    const float* fc_b    = (const float*)d_in[2];
    const float* conv1_w = (const float*)d_in[3];
    const float* conv1_b = (const float*)d_in[4];
    const float* conv2_w = (const float*)d_in[5];
    const float* conv2_b = (const float*)d_in[6];
    float* out = (float*)d_out;

    // B=8, 64 patch rows, 4 groups of 16 patches per row -> 2048 blocks
    const int nblocks = 8 * 64 * 4;
    ciam_fused_kernel<<<nblocks, 256, 0, stream>>>(
        x, fc_w, fc_b, conv1_w, conv1_b, conv2_w, conv2_b, out);
}